// MultiHeadedAttentionRel_58531814310503
// MI455X (gfx1250) — compile-verified
//
#include <hip/hip_runtime.h>

typedef float v2f __attribute__((ext_vector_type(2)));
typedef float v8f __attribute__((ext_vector_type(8)));
typedef unsigned int u32x4 __attribute__((ext_vector_type(4)));
typedef int i32x4 __attribute__((ext_vector_type(4)));
typedef int i32x8 __attribute__((ext_vector_type(8)));

#define SLEN 1024
#define DMODEL 1024
#define NHEAD 16
#define DKH 64

// ---------------------------------------------------------------------------
// Kernel 1: positional gather + Q/K projection (fp32 WMMA 16x16x4).
// grid = (M/128, D/64, 2)  z=0 -> Q (Wq,bq), z=1 -> K (Wk,bk)
// Output layout: [B, H, S, DK]
// B tile stored pair-interleaved in LDS: (k,n) -> [(k>>1)*160 + n*2 + (k&1)]
// so each WMMA B-fragment is one aligned ds_load_b64.
// ---------------------------------------------------------------------------
#define BP 160   // pair-row stride (dwords); 160 % 64 == 32 -> halves on disjoint banks

__global__ __launch_bounds__(256) void gather_proj_wmma(
    const int*   __restrict__ posMat,   // [B*S]
    const float* __restrict__ pos_enc,  // [POS_VOCAB, D]
    const float* __restrict__ Wq, const float* __restrict__ bq,
    const float* __restrict__ Wk, const float* __restrict__ bk,
    float* __restrict__ Qout, float* __restrict__ Kout)
{
    __shared__ float Alds[128 * 36];   // 128 rows x 32 k, stride 36 (conflict-free b64)
    __shared__ float Blds[16 * BP];    // 16 k-pairs x 64 n x 2
    __shared__ int   posLds[128];

    const int tid = threadIdx.x;
    const int r0  = blockIdx.x * 128;
    const int n0  = blockIdx.y * 64;
    const bool isK = (blockIdx.z != 0);
    const float* W   = isK ? Wk : Wq;
    const float* bv  = isK ? bk : bq;
    float*       Out = isK ? Kout : Qout;

    if (tid < 128) posLds[tid] = posMat[r0 + tid];
    __syncthreads();

    const int wave = tid >> 5;
    const int lane = tid & 31;
    const int half = lane >> 4;
    const int mrow = lane & 15;

    v8f acc[4] = {};

    for (int kc = 0; kc < DMODEL; kc += 32) {
        // --- A tile: gathered rows of pos_encoding, coalesced float4 along k
#pragma unroll
        for (int i = 0; i < 4; ++i) {
            int idx = tid + i * 256;
            int row = idx >> 3;
            int kq  = (idx & 7) << 2;
            float4 v = *(const float4*)(pos_enc + (size_t)posLds[row] * DMODEL + kc + kq);
            *(float4*)&Alds[row * 36 + kq] = v;
        }
        // --- B tile: B[k][n] = W[n0+n][kc+k], pair-interleaved
#pragma unroll
        for (int i = 0; i < 2; ++i) {
            int idx = tid + i * 256;
            int n  = idx >> 3;
            int kq = (idx & 7) << 2;
            float4 v = *(const float4*)(W + (size_t)(n0 + n) * DMODEL + kc + kq);
            v2f lo = {v.x, v.y};
            v2f hi = {v.z, v.w};
            *(v2f*)&Blds[((kq >> 1) + 0) * BP + n * 2] = lo;
            *(v2f*)&Blds[((kq >> 1) + 1) * BP + n * 2] = hi;
        }
        __syncthreads();

#pragma unroll
        for (int kk = 0; kk < 8; ++kk) {
            v2f a = *(const v2f*)&Alds[(wave * 16 + mrow) * 36 + kk * 4 + half * 2];
            const int pb = (kk * 2 + half) * BP + mrow * 2;
#pragma unroll
            for (int nt = 0; nt < 4; ++nt) {
                v2f b = *(const v2f*)&Blds[pb + nt * 32];
                acc[nt] = __builtin_amdgcn_wmma_f32_16x16x4_f32(
                    false, a, false, b, (short)0, acc[nt], false, false);
            }
        }
        __syncthreads();
    }

    // Epilogue: add bias vector, scatter to [B,H,S,DK]
#pragma unroll
    for (int nt = 0; nt < 4; ++nt) {
        const int colg = n0 + nt * 16 + mrow;
        const float badd = bv[colg];
        const int h  = colg >> 6;
        const int dk = colg & 63;
#pragma unroll
        for (int j = 0; j < 8; ++j) {
            const int m = r0 + wave * 16 + half * 8 + j;
            const int bidx = m >> 10;
            const int s    = m & 1023;
            const size_t o = (((size_t)(bidx * NHEAD + h) * SLEN) + s) * DKH + dk;
            Out[o] = acc[nt][j] + badd;
        }
    }
}

// ---------------------------------------------------------------------------
// Kernel 2: scores[bh] = (Q[bh] @ K[bh]^T) / sqrt(128) + bias[bh]
// grid = (S/128, S/128, B*H); 128x128 tile per block, 8 waves.
// Q tile (128x64 fp32) staged by the Tensor Data Mover with LDS padding
// (64 dwords data + 4 dwords pad -> row stride 68, conflict-free A-fragments).
// K tile staged manually (transposed, pair-interleaved) in two 32-wide chunks.
// ---------------------------------------------------------------------------
#define KP 288   // pair-row stride (dwords); 288 % 64 == 32

__global__ __launch_bounds__(256) void scores_wmma(
    const float* __restrict__ Q,     // [B*H, S, DK]
    const float* __restrict__ Km,    // [B*H, S, DK]
    const float* __restrict__ bias,  // [B*H, S, S]
    float* __restrict__ out)
{
    __shared__ float Qlds[128 * 68];   // TDM destination, padded rows
    __shared__ float Klds[16 * KP];    // 16 d-pairs x 128 n x 2

    const int tid = threadIdx.x;
    const int q0  = blockIdx.x * 128;
    const int k0  = blockIdx.y * 128;
    const int bh  = blockIdx.z;
    const size_t qkbase = (size_t)bh * SLEN * DKH;

    const int wave = tid >> 5;
    const int lane = tid & 31;
    const int half = lane >> 4;
    const int mrow = lane & 15;

    // ---- TDM: async load of the whole 128x64 Q tile into padded LDS --------
    if (wave == 0) {
        unsigned long long gaddr =
            (unsigned long long)(uintptr_t)(Q + qkbase + (size_t)q0 * DKH);
        unsigned lds_base = (unsigned)(uintptr_t)(&Qlds[0]);   // LDS byte offset

        u32x4 g0;
        g0.x = 1u;                                   // count=1, no gather
        g0.y = lds_base;                             // lds_addr
        g0.z = (unsigned)(gaddr & 0xFFFFFFFFull);    // global_addr[31:0]
        g0.w = (unsigned)(gaddr >> 32) | 0x80000000u; // global_addr[56:32] | type=2

        i32x8 g1;
        g1[0] = 0x07520000;        // data_size=4B | pad_enable | interval=64dw | amount=4dw
        g1[1] = (int)(64u << 16);  // tensor_dim0[15:0]=64 (abar_addr=0)
        g1[2] = (int)(128u << 16); // tensor_dim1[15:0]=128 (dim0 hi=0)
        g1[3] = (int)(64u << 16);  // tile_dim0=64 (dim1 hi=0)
        g1[4] = 128;               // tile_dim1=128, tile_dim2=0
        g1[5] = 64;                // tensor_dim0_stride = 64 elements
        g1[6] = 0;
        g1[7] = 0;

        i32x4 gz4 = {};            // groups 2/3: dims/strides zero (unused)
        i32x8 gz8 = {};            // trailing descriptor group (unused)
        __builtin_amdgcn_tensor_load_to_lds(g0, g1, gz4, gz4, gz8, 0);
    }

    v8f acc[8] = {};

    for (int c = 0; c < 2; ++c) {              // DK = 64 in two 32-wide chunks
        const int d0 = c * 32;
        // K tile chunk: transpose + pair-interleave
#pragma unroll
        for (int i = 0; i < 4; ++i) {
            int idx = tid + i * 256;
            int n  = idx >> 3;
            int ld = (idx & 7) << 2;
            float4 v = *(const float4*)(Km + qkbase + (size_t)(k0 + n) * DKH + d0 + ld);
            v2f lo = {v.x, v.y};
            v2f hi = {v.z, v.w};
            *(v2f*)&Klds[((ld >> 1) + 0) * KP + n * 2] = lo;
            *(v2f*)&Klds[((ld >> 1) + 1) * KP + n * 2] = hi;
        }
        if (c == 0 && wave == 0) {
            __builtin_amdgcn_s_wait_tensorcnt(0);  // TDM done before barrier
        }
        __syncthreads();

#pragma unroll
        for (int dd = 0; dd < 8; ++dd) {
            v2f a = *(const v2f*)&Qlds[(wave * 16 + mrow) * 68 + d0 + dd * 4 + half * 2];
            const int pb = (dd * 2 + half) * KP + mrow * 2;
#pragma unroll
            for (int nt = 0; nt < 8; ++nt) {
                v2f b = *(const v2f*)&Klds[pb + nt * 32];
                acc[nt] = __builtin_amdgcn_wmma_f32_16x16x4_f32(
                    false, a, false, b, (short)0, acc[nt], false, false);
            }
        }
        __syncthreads();
    }

    const float scale = 0.08838834764831845f;  // 1/sqrt(2*DK) = 1/sqrt(128)
#pragma unroll
    for (int nt = 0; nt < 8; ++nt) {
        const int kcol = k0 + nt * 16 + mrow;
#pragma unroll
        for (int j = 0; j < 8; ++j) {
            const int qrow = q0 + wave * 16 + half * 8 + j;
            const size_t o = ((size_t)bh * SLEN + qrow) * SLEN + kcol;
            out[o] = acc[nt][j] * scale + bias[o];
        }
    }
}

// ---------------------------------------------------------------------------
extern "C" void kernel_launch(void* const* d_in, const int* in_sizes, int n_in,
                              void* d_out, int out_size, void* d_ws, size_t ws_size,
                              hipStream_t stream) {
    // inputs: 0:query 1:key 2:value (unused by reference)
    //         3:positionMat 4:bias 5:pos_encoding 6:Wq 7:bq 8:Wk 9:bk
    const int*   posMat = (const int*)  d_in[3];
    const float* bias   = (const float*)d_in[4];
    const float* pe     = (const float*)d_in[5];
    const float* Wq     = (const float*)d_in[6];
    const float* bq     = (const float*)d_in[7];
    const float* Wk     = (const float*)d_in[8];
    const float* bk     = (const float*)d_in[9];

    float* Qws = (float*)d_ws;                          // 16 MB
    float* Kws = Qws + (size_t)4 * SLEN * DMODEL;       // +16 MB
    float* out = (float*)d_out;

    dim3 g1(4 * SLEN / 128, DMODEL / 64, 2);            // (32,16,2)
    gather_proj_wmma<<<g1, 256, 0, stream>>>(posMat, pe, Wq, bq, Wk, bk, Qws, Kws);

    dim3 g2(SLEN / 128, SLEN / 128, 4 * NHEAD);         // (8,8,64)
    scores_wmma<<<g2, 256, 0, stream>>>(Qws, Kws, bias, out);
}